// BSplineACTF_52956946760170
// MI455X (gfx1250) — compile-verified
//
#include <hip/hip_runtime.h>
#include <hip/hip_bf16.h>
#include <stdint.h>

// ---------------------------------------------------------------------------
// B-spline activation (KAN-style), MI455X / gfx1250.
//
// Memory-bound: 256 MB of traffic -> ~11 us floor @ 23.3 TB/s.
// Uniform knots => per (channel, interval) the spline is a fixed cubic:
//   out = ((C3*t + C2)*t + C1)*t + C0,  t = frac((x - g0)/h), j = floor(...)
// Coefficient table (32 ch x 11 intervals x float4 = 5.6 KB) built once per
// block in LDS from W. x is streamed HBM -> LDS via the Tensor Data Mover
// (tensor_load_to_lds, double-buffered per wave, s_wait_tensorcnt), computed
// with ds_read_b128 + ~10 VALU ops/element, stored with global_store_b128.
// ---------------------------------------------------------------------------

typedef __attribute__((ext_vector_type(4))) unsigned int v4u;
typedef __attribute__((ext_vector_type(8))) int          v8i;
typedef __attribute__((ext_vector_type(4))) int          v4i;

#define WAVES        8
#define SEG          512                   // floats per wave per TDM chunk
#define PASSES       (SEG / 128)           // float4s per lane per chunk (4)
#define ITERS        4
#define BLOCK_ELEMS  (WAVES * SEG)         // 4096 floats per block-iter
#define COEFF_BYTES  8192                  // 352 * 16 = 5632 used, padded
#define XBUF_BYTES   (SEG * 4)             // 2048 B per wave per buffer
#define BUF_STRIDE   (WAVES * XBUF_BYTES)  // 16384 B per buffer
#define SMEM_BYTES   (COEFF_BYTES + 2 * BUF_STRIDE)  // 40960 B

// Issue one TDM descriptor: copy `nfloats` contiguous f32 from global `gsrc`
// to LDS byte offset `lds_off` (1-row 2D tile, data_size = 4 bytes).
__device__ __forceinline__ void tdm_load_row(const float* gsrc,
                                             unsigned lds_off, int nfloats) {
  uint64_t ga = (uint64_t)(uintptr_t)gsrc;
  v4u g0;
  g0[0] = 1u;                                         // count=1, user mode
  g0[1] = lds_off;                                    // lds_addr (bytes)
  g0[2] = (unsigned)(ga & 0xFFFFFFFFu);               // global_addr[31:0]
  g0[3] = (unsigned)((ga >> 32) & 0x1FFFFFFu)         // global_addr[56:32]
        | (2u << 30);                                 // type = 2 ("image")
  v8i g1;
  g1[0] = (int)(2u << 16);                            // data_size = 4B
  g1[1] = (int)(((unsigned)nfloats & 0xFFFFu) << 16); // tensor_dim0[15:0]
  g1[2] = (int)(1u << 16);                            // dim0 hi=0; tensor_dim1=1
  g1[3] = (int)(((unsigned)nfloats & 0xFFFFu) << 16); // dim1 hi=0; tile_dim0
  g1[4] = 1;                                          // tile_dim1=1, tile_dim2=0
  g1[5] = nfloats;                                    // tensor_dim0_stride lo32
  g1[6] = 0;                                          // stride hi / dim1_stride lo
  g1[7] = 0;                                          // dim1_stride hi
  v4i z4 = {0, 0, 0, 0};
  v8i z8 = {0, 0, 0, 0, 0, 0, 0, 0};
  // amdgpu-toolchain (clang-23) 6-arg form.
  __builtin_amdgcn_tensor_load_to_lds(g0, g1, z4, z4, z8, 0);
}

__device__ __forceinline__ float eval1(float xv, float invh, float bias,
                                       const float4* coeff, int crow) {
  float s  = __builtin_fmaf(xv, invh, bias);   // (x - g0)/h
  float jf = floorf(s);
  float t  = s - jf;
  int   j  = (int)jf;
  // j outside [0,10] reads a neighbouring row or stray LDS; result masked below anyway.
  float4 C = coeff[crow + j];
  float r  = __builtin_fmaf(
                 __builtin_fmaf(__builtin_fmaf(C.w, t, C.z), t, C.y), t, C.x);
  // s in [0, 11)  <=>  (unsigned)floor(s) < 11 : one v_cmp_lt_u32.
  return ((unsigned)j < 11u) ? r : 0.0f;
}

// C = cubic power-basis coefficients for channel c, interval j.
__device__ __forceinline__ void write_coeff(float4* coeff,
                                            const float* __restrict__ W,
                                            int c, int j) {
  float w[4];
#pragma unroll
  for (int i = 0; i < 4; ++i) {
    int k = j - 3 + i;                       // basis index; zero if OOR
    w[i] = (k >= 0 && k < 8) ? W[c * 8 + k] : 0.0f;
  }
  const float s6 = 1.0f / 6.0f;
  float4 C;
  C.x = (w[0] + 4.0f * w[1] + w[2]) * s6;            // t^0
  C.y = (w[2] - w[0]) * 0.5f;                        // t^1
  C.z = (w[0] + w[2]) * 0.5f - w[1];                 // t^2
  C.w = (w[3] - w[0]) * s6 + (w[1] - w[2]) * 0.5f;   // t^3
  coeff[c * 11 + j] = C;
}

__global__ __launch_bounds__(256) void bspline_act_kernel(
    const float* __restrict__ x, const float* __restrict__ grid,
    const float* __restrict__ W, float* __restrict__ out) {
  __shared__ __align__(16) char smem[SMEM_BYTES];
  float4* coeff = (float4*)smem;

  // Uniform-grid parameters (scalar loads; uniform across block).
  float g0   = grid[0];
  float invh = 1.0f / (grid[1] - grid[0]);
  float bias = -g0 * invh;

  // Build per-(channel, interval) cubic coefficients from W.
  // 32 channels x 11 intervals; 256 threads cover j=0..7 in one slab,
  // threads 0..95 cover j=8..10. Pure shift/mask indexing (no div/mod).
  {
    int tid = threadIdx.x;
    write_coeff(coeff, W, tid & 31, tid >> 5);              // j = 0..7
    if (tid < 96)
      write_coeff(coeff, W, tid & 31, 8 + (tid >> 5));      // j = 8..10
  }

  const int lane = threadIdx.x & 31;
  const int wave = threadIdx.x >> 5;
  const int cb   = (lane & 7) * 4;  // channel base of this lane's float4
  const int crow0 = (cb + 0) * 11, crow1 = (cb + 1) * 11;
  const int crow2 = (cb + 2) * 11, crow3 = (cb + 3) * 11;

  const size_t   blockBase = (size_t)blockIdx.x * (BLOCK_ELEMS * ITERS);
  const unsigned ldsbase   = COEFF_BYTES + (unsigned)wave * XBUF_BYTES;

  // Prologue: each wave DMA-loads its iter-0 segment into buffer 0.
  tdm_load_row(x + blockBase + wave * SEG, ldsbase + 0 * BUF_STRIDE, SEG);

  __syncthreads();  // coefficient table visible to all waves

#pragma unroll
  for (int it = 0; it < ITERS; ++it) {
    if (it + 1 < ITERS) {
      tdm_load_row(x + blockBase + (size_t)(it + 1) * BLOCK_ELEMS + wave * SEG,
                   ldsbase + (unsigned)((it + 1) & 1) * BUF_STRIDE, SEG);
      __builtin_amdgcn_s_wait_tensorcnt(1);  // chunk `it` has landed
    } else {
      __builtin_amdgcn_s_wait_tensorcnt(0);  // last chunk has landed
    }
    __asm__ volatile("" ::: "memory");       // keep LDS reads below the wait

    const float4* xb =
        (const float4*)(smem + ldsbase + (unsigned)(it & 1) * BUF_STRIDE);
    const size_t ebase = blockBase + (size_t)it * BLOCK_ELEMS + wave * SEG;

#pragma unroll
    for (int p = 0; p < PASSES; ++p) {
      int q = p * 32 + lane;
      float4 xv = xb[q];                     // ds_read_b128
      float4 r;
      r.x = eval1(xv.x, invh, bias, coeff, crow0);
      r.y = eval1(xv.y, invh, bias, coeff, crow1);
      r.z = eval1(xv.z, invh, bias, coeff, crow2);
      r.w = eval1(xv.w, invh, bias, coeff, crow3);
      *(float4*)(out + ebase + (size_t)q * 4) = r;     // global_store_b128
    }
  }
}

extern "C" void kernel_launch(void* const* d_in, const int* in_sizes, int n_in,
                              void* d_out, int out_size, void* d_ws,
                              size_t ws_size, hipStream_t stream) {
  const float* x    = (const float*)d_in[0];
  const float* grid = (const float*)d_in[1];
  const float* W    = (const float*)d_in[2];
  float*       out  = (float*)d_out;

  int n      = in_sizes[0];                       // 33,554,432
  int blocks = n / (BLOCK_ELEMS * ITERS);         // 2048
  hipLaunchKernelGGL(bspline_act_kernel, dim3(blocks), dim3(256), 0, stream,
                     x, grid, W, out);
}